// YmirYolov5_49924699849378
// MI455X (gfx1250) — compile-verified
//
#include <hip/hip_runtime.h>
#include <hip/hip_bf16.h>
#include <cstdint>

#define NANCH 100800
#define NCLS 80
#define ROWF 85              // floats per anchor row
#define TILE 128             // anchors per stage-A block
#define KPRE 1024
#define CAP 2048
#define MAXDET 100
#define CONF_T 0.25f
#define IOU_T 0.45f
#define MAX_WH 7680.0f
#define EPSF 1e-7f

#if __has_builtin(__builtin_amdgcn_global_load_async_to_lds_b128) && \
    __has_builtin(__builtin_amdgcn_s_wait_asynccnt)
#define USE_ASYNC_LDS 1
#else
#define USE_ASYNC_LDS 0
#endif

typedef int v4i __attribute__((vector_size(16)));
typedef __attribute__((address_space(1))) v4i* gv4i_p;   // global
typedef __attribute__((address_space(3))) v4i* lv4i_p;   // LDS

__device__ __forceinline__ unsigned flip_key(float f) {
  unsigned u = __float_as_uint(f);
  // order-preserving map: ascending uint == ascending float (incl. negatives)
  return u ^ (((int)u >> 31) | 0x80000000u);
}

// ---------------- Stage A: streaming score reduction (bandwidth bound) ----
__global__ __launch_bounds__(TILE) void k_score(const float* __restrict__ pred,
                                                float* __restrict__ scores,
                                                int nAnch) {
  __shared__ float tileBuf[TILE * ROWF];   // 43,520 B
  const int b = blockIdx.y;
  const int row0 = blockIdx.x * TILE;
  const int rows = min(TILE, nAnch - row0);
  if (rows <= 0) return;
  const float* src = pred + ((size_t)b * nAnch + row0) * ROWF;
  // contiguous span rows*340 bytes, 16B aligned (rows % 4 == 0 by construction)
  const int nchunk = (rows * ROWF) >> 2;   // 16-byte chunks
  for (int c = threadIdx.x; c < nchunk; c += blockDim.x) {
#if USE_ASYNC_LDS
    __builtin_amdgcn_global_load_async_to_lds_b128(
        (gv4i_p)(uintptr_t)(src + 4 * c),
        (lv4i_p)(uintptr_t)(&tileBuf[4 * c]),
        0, 0);
#else
    ((float4*)tileBuf)[c] = ((const float4*)src)[c];
#endif
  }
#if USE_ASYNC_LDS
  __builtin_amdgcn_s_wait_asynccnt(0);
#endif
  __syncthreads();

  const int t = threadIdx.x;
  if (t < rows) {
    const float* rp = &tileBuf[t * ROWF];  // stride 85 words: conflict-free (gcd(85,64)=1)
    float obj = rp[4];
    float best = rp[5] * obj;
    for (int c = 1; c < NCLS; ++c) {
      float v = rp[5 + c] * obj;
      if (v > best) best = v;
    }
    bool valid = (obj > CONF_T) && (best > CONF_T);
    scores[(size_t)b * nAnch + row0 + t] = valid ? best : -1.0f;
  }
}

// ---------------- Stage B: exact top-1024 per batch (radix select + sort) --
__global__ __launch_bounds__(1024) void k_select(const float* __restrict__ scores,
                                                 int* __restrict__ selIdx,
                                                 float* __restrict__ selScore,
                                                 int nAnch) {
  __shared__ int hist[256];
  __shared__ unsigned long long cand[CAP];
  __shared__ int s_cnt;
  __shared__ unsigned s_prefix;
  __shared__ int s_wanted;
  const int b = blockIdx.x;
  const int t = threadIdx.x;
  const float* sc = scores + (size_t)b * nAnch;

  if (t == 0) { s_prefix = 0u; s_wanted = KPRE; }

  for (int s = 24; s >= 0; s -= 8) {
    if (t < 256) hist[t] = 0;
    __syncthreads();
    unsigned prefix = s_prefix;
    for (int i = t; i < nAnch; i += 1024) {
      unsigned key = flip_key(sc[i]);
      unsigned sh = (unsigned)(s + 8);
      bool ok = (sh >= 32u) || ((key >> sh) == (prefix >> sh));
      if (ok) atomicAdd(&hist[(key >> s) & 255u], 1);
    }
    __syncthreads();
    if (t == 0) {
      int wanted = s_wanted, cum = 0, bin = 0;
      for (int i2 = 255; i2 >= 0; --i2) {
        int c = hist[i2];
        if (cum + c >= wanted) { bin = i2; s_wanted = wanted - cum; break; }
        cum += c;
      }
      s_prefix = prefix | ((unsigned)bin << s);
    }
    __syncthreads();
  }

  const unsigned T = s_prefix;   // exact 1024-th largest key
  if (t == 0) s_cnt = 0;
  __syncthreads();
  for (int i = t; i < nAnch; i += 1024) {
    unsigned key = flip_key(sc[i]);
    if (key >= T) {
      int pos = atomicAdd(&s_cnt, 1);
      if (pos < CAP)
        cand[pos] = ((unsigned long long)(~key) << 32) | (unsigned)i;
    }
  }
  __syncthreads();
  const int cnt = min(s_cnt, CAP);
  for (int i = t; i < CAP; i += 1024)
    if (i >= cnt) cand[i] = ~0ull;       // sorts to the end
  __syncthreads();

  // ascending bitonic sort of (~key, idx): gives score desc, idx asc on ties
  for (int k = 2; k <= CAP; k <<= 1) {
    for (int j = k >> 1; j > 0; j >>= 1) {
      for (int i = t; i < CAP; i += 1024) {
        int l = i ^ j;
        if (l > i) {
          unsigned long long a = cand[i], c2 = cand[l];
          bool up = ((i & k) == 0);
          if (up ? (a > c2) : (a < c2)) { cand[i] = c2; cand[l] = a; }
        }
      }
      __syncthreads();
    }
  }

  if (t < KPRE) {
    unsigned long long v = cand[t];
    unsigned idx = (unsigned)v;
    int o = b * KPRE + t;
    if (t < cnt && idx < (unsigned)nAnch) {
      selIdx[o] = (int)idx;
      selScore[o] = sc[idx];
    } else {
      selIdx[o] = 0;
      selScore[o] = -1.0f;
    }
  }
}

// ---------------- Stage C: gather + greedy NMS + top-100 emit -------------
__global__ __launch_bounds__(1024) void k_nms(const float* __restrict__ pred,
                                              const int* __restrict__ selIdx,
                                              const float* __restrict__ selScore,
                                              float* __restrict__ out,
                                              int nAnch) {
  __shared__ float ox1[KPRE], oy1[KPRE], ox2[KPRE], oy2[KPRE];   // class-offset boxes
  __shared__ float px1[KPRE], py1[KPRE], px2[KPRE], py2[KPRE];   // plain boxes
  __shared__ float sarea[KPRE], sscore[KPRE], sclsf[KPRE];
  __shared__ int skeep[KPRE], srank[KPRE];
  __shared__ int sowner[MAXDET];

  const int b = blockIdx.x, t = threadIdx.x;
  const int idx = selIdx[b * KPRE + t];
  const float score = selScore[b * KPRE + t];
  const float* rp = pred + ((size_t)b * nAnch + idx) * ROWF;

  float cx = rp[0], cy = rp[1], w = rp[2], h = rp[3], obj = rp[4];
  int cls = 0;
  float best = rp[5] * obj;
  for (int c = 1; c < NCLS; ++c) {
    float v = rp[5 + c] * obj;
    if (v > best) { best = v; cls = c; }
  }
  float x1 = cx - w * 0.5f, y1v = cy - h * 0.5f;
  float x2 = cx + w * 0.5f, y2v = cy + h * 0.5f;
  float off = (float)cls * MAX_WH;
  float mX1 = x1 + off, mY1 = y1v + off, mX2 = x2 + off, mY2 = y2v + off;
  float mArea = (x2 - x1) * (y2v - y1v);

  ox1[t] = mX1; oy1[t] = mY1; ox2[t] = mX2; oy2[t] = mY2;
  px1[t] = x1; py1[t] = y1v; px2[t] = x2; py2[t] = y2v;
  sarea[t] = mArea; sscore[t] = score; sclsf[t] = (float)cls;
  int myKeep = (score > 0.0f) ? 1 : 0;
  skeep[t] = myKeep;
  __syncthreads();

  // forward greedy NMS (identical to the reference scan)
  for (int i = 0; i < KPRE; ++i) {
    if (skeep[i]) {
      if (t > i && myKeep) {
        float ix1 = fmaxf(mX1, ox1[i]);
        float iy1 = fmaxf(mY1, oy1[i]);
        float ix2 = fminf(mX2, ox2[i]);
        float iy2 = fminf(mY2, oy2[i]);
        float iw = fmaxf(ix2 - ix1, 0.0f);
        float ih = fmaxf(iy2 - iy1, 0.0f);
        float inter = iw * ih;
        float uni = mArea + sarea[i] - inter + EPSF;
        if (inter / uni > IOU_T) { myKeep = 0; skeep[t] = 0; }
      }
    }
    __syncthreads();
  }

  // inclusive scan of keep flags -> compaction ranks
  srank[t] = myKeep;
  __syncthreads();
  for (int o2 = 1; o2 < KPRE; o2 <<= 1) {
    int u = (t >= o2) ? srank[t - o2] : 0;
    __syncthreads();
    srank[t] += u;
    __syncthreads();
  }
  int rex = srank[t] - myKeep;

  if (t < MAXDET) sowner[t] = -1;
  __syncthreads();
  if (myKeep && rex < MAXDET) sowner[rex] = t;
  __syncthreads();

  if (t < MAXDET) {
    float* o = out + ((size_t)b * MAXDET + t) * 6;
    int ow = sowner[t];
    if (ow >= 0) {
      o[0] = px1[ow]; o[1] = py1[ow]; o[2] = px2[ow]; o[3] = py2[ow];
      o[4] = sscore[ow]; o[5] = sclsf[ow];
    } else {
      o[0] = 0.f; o[1] = 0.f; o[2] = 0.f; o[3] = 0.f; o[4] = 0.f; o[5] = 0.f;
    }
  }
}

extern "C" void kernel_launch(void* const* d_in, const int* in_sizes, int n_in,
                              void* d_out, int out_size, void* d_ws, size_t ws_size,
                              hipStream_t stream) {
  const float* pred = (const float*)d_in[0];
  float* out = (float*)d_out;
  const int B = in_sizes[0] / (NANCH * ROWF);

  char* ws = (char*)d_ws;
  size_t offScores = 0;
  size_t offSelIdx = (offScores + (size_t)B * NANCH * sizeof(float) + 255) & ~(size_t)255;
  size_t offSelScore = (offSelIdx + (size_t)B * KPRE * sizeof(int) + 255) & ~(size_t)255;
  float* scores = (float*)(ws + offScores);
  int* selIdx = (int*)(ws + offSelIdx);
  float* selScore = (float*)(ws + offSelScore);

  dim3 gA((NANCH + TILE - 1) / TILE, B);
  k_score<<<gA, TILE, 0, stream>>>(pred, scores, NANCH);
  k_select<<<B, 1024, 0, stream>>>(scores, selIdx, selScore, NANCH);
  k_nms<<<B, 1024, 0, stream>>>(pred, selIdx, selScore, out, NANCH);
}